// Net_35579509080699
// MI455X (gfx1250) — compile-verified
//
#include <hip/hip_runtime.h>
#include <math.h>

#define NV 8192
#define NE 49152
#define MD 25

__device__ __constant__ float cLC = -0.22703196f;
__device__ __constant__ float cUC =  0.36853024f;
__device__ __constant__ float cLM =  1.2585511f;
__device__ __constant__ float cUM =  1.648841f;

typedef float v2f __attribute__((ext_vector_type(2)));
typedef float v8f __attribute__((ext_vector_type(8)));

// ---------------- setup kernels ----------------

__global__ __launch_bounds__(256) void zero_kernel(int* counts, int* cursor) {
    int i = blockIdx.x * 256 + threadIdx.x;
    if (i < NV) counts[i] = 0;
    else if (i < 2 * NV) cursor[i - NV] = 0;
}

// transform(x) -> h0, spline basis/indices, dst histogram
__global__ __launch_bounds__(256) void prep_kernel(const float* __restrict__ x,
                                                   const int* __restrict__ ei,
                                                   const float* __restrict__ pseudo,
                                                   float* __restrict__ basis,
                                                   int* __restrict__ wi,
                                                   float* __restrict__ h0,
                                                   int* __restrict__ counts) {
    int e = blockIdx.x * 256 + threadIdx.x;
    if (e < NV) {
        const float R = 10.0f;
        float t0 = (x[2 * e + 0] - cLC) / (cUC - cLC) * (2.0f * R) - R;
        float t1 = (x[2 * e + 1] - cLM) / (cUM - cLM) * (2.0f * R) - R;
        h0[e * 64 + 0] = fminf(fmaxf(t0, -R), R);
        h0[e * 64 + 1] = fminf(fmaxf(t1, -R), R);
    }
    if (e >= NE) return;

    float fr[3]; int lo[3];
    #pragma unroll
    for (int d = 0; d < 3; d++) {
        float v = pseudo[e * 3 + d] * (float)(MD - 1);
        float fl = floorf(v);
        lo[d] = (int)fl;
        fr[d] = v - fl;
    }
    int dst = ei[NE + e];
    atomicAdd(&counts[dst], 1);
    #pragma unroll
    for (int s = 0; s < 8; s++) {
        float w = 1.0f; int idx = 0; int stride = 1;
        #pragma unroll
        for (int d = 0; d < 3; d++) {
            int b = (s >> d) & 1;
            w *= b ? fr[d] : (1.0f - fr[d]);
            idx += (lo[d] + b) * stride;
            stride *= MD;
        }
        basis[e * 8 + s] = w;
        wi[e * 8 + s] = idx;
    }
}

// exclusive scan of 8192 counts with one block
__global__ __launch_bounds__(256) void scan_kernel(const int* __restrict__ counts,
                                                   int* __restrict__ off) {
    __shared__ int ssum[256];
    __shared__ int spre[257];
    int t = threadIdx.x;
    int base = t * 32;
    int loc[32];
    int s = 0;
    #pragma unroll
    for (int i = 0; i < 32; i++) { loc[i] = s; s += counts[base + i]; }
    ssum[t] = s;
    __syncthreads();
    if (t == 0) {
        int r = 0;
        for (int i = 0; i < 256; i++) { spre[i] = r; r += ssum[i]; }
        spre[256] = r;
    }
    __syncthreads();
    int b = spre[t];
    #pragma unroll
    for (int i = 0; i < 32; i++) off[base + i] = b + loc[i];
    if (t == 0) off[NV] = spre[256];
}

__global__ __launch_bounds__(256) void fill_csr_kernel(const int* __restrict__ ei,
                                                       const int* __restrict__ off,
                                                       int* __restrict__ cursor,
                                                       int* __restrict__ eid) {
    int e = blockIdx.x * 256 + threadIdx.x;
    if (e >= NE) return;
    int dst = ei[NE + e];
    int pos = off[dst] + atomicAdd(&cursor[dst], 1);
    eid[pos] = e;
}

// canonicalize per-vertex edge order -> deterministic float sums
__global__ __launch_bounds__(256) void sort_csr_kernel(const int* __restrict__ off,
                                                       int* __restrict__ eid) {
    int v = blockIdx.x * 256 + threadIdx.x;
    if (v >= NV) return;
    int a = off[v], b = off[v + 1];
    for (int i = a + 1; i < b; i++) {
        int key = eid[i];
        int j = i - 1;
        while (j >= a && eid[j] > key) { eid[j + 1] = eid[j]; j--; }
        eid[j + 1] = key;
    }
}

// ---------------- per-layer kernels ----------------

// one wave per edge, lanes across Fout; fixed s order => deterministic
template <int Fin, int Fout>
__global__ __launch_bounds__(256) void edge_msg_kernel(const float* __restrict__ h,
                                                       const float* __restrict__ W,
                                                       const int* __restrict__ ei,
                                                       const float* __restrict__ basis,
                                                       const int* __restrict__ wi,
                                                       float* __restrict__ msg) {
    int warp = threadIdx.x >> 5, lane = threadIdx.x & 31;
    int e = blockIdx.x * 8 + warp;
    int src = ei[e];
    float x0 = h[src * 64 + lane];
    float x1 = (Fin > 32) ? h[src * 64 + 32 + lane] : 0.0f;
    float acc0 = 0.0f, acc1 = 0.0f;
    #pragma unroll
    for (int s = 0; s < 8; s++) {
        int k = wi[e * 8 + s];
        float bs = basis[e * 8 + s];
        const float* Wk = W + (size_t)k * (Fin * Fout);
        #pragma unroll
        for (int i = 0; i < Fin; i++) {
            float xv = (i < 32) ? __shfl(x0, i, 32) : __shfl(x1, i - 32, 32);
            float t = bs * xv;
            if constexpr (Fout == 64) {
                float2 w = ((const float2*)(Wk + i * 64))[lane];
                acc0 = fmaf(t, w.x, acc0);
                acc1 = fmaf(t, w.y, acc1);
            } else {
                acc0 = fmaf(t, Wk[i * Fout + lane], acc0);
            }
        }
    }
    if constexpr (Fout == 64) {
        ((float2*)(msg + (size_t)e * 64))[lane] = make_float2(acc0, acc1);
    } else {
        msg[(size_t)e * 64 + lane] = acc0;
    }
}

// dense root GEMM tmp = h @ root via V_WMMA_F32_16X16X4_F32; one wave = 16x16 C tile
template <int Fin, int Fout>
__global__ __launch_bounds__(256) void wmma_root_kernel(const float* __restrict__ h,
                                                        const float* __restrict__ root,
                                                        float* __restrict__ tmp) {
    int warp = threadIdx.x >> 5, lane = threadIdx.x & 31;
    constexpr int TN = Fout / 16;
    int tile = blockIdx.x * 8 + warp;
    int m = (tile / TN) * 16;
    int n = (tile % TN) * 16;
    int hi = lane >> 4;     // half-wave: K offset 0 or 2
    int lm = lane & 15;     // row (A) / col (B,C)
    v8f c = {};
    #pragma unroll
    for (int k = 0; k < Fin; k += 4) {
        int ka = k + 2 * hi;
        v2f a = { h[(m + lm) * 64 + ka], h[(m + lm) * 64 + ka + 1] };
        v2f b = { root[ka * Fout + n + lm], root[(ka + 1) * Fout + n + lm] };
        c = __builtin_amdgcn_wmma_f32_16x16x4_f32(false, a, false, b, (short)0, c,
                                                  false, false);
    }
    #pragma unroll
    for (int j = 0; j < 8; j++)
        tmp[(m + j + 8 * hi) * 64 + n + lm] = c[j];
}

// layer-1 root (Fin=2): trivial
__global__ __launch_bounds__(256) void root2_kernel(const float* __restrict__ h,
                                                    const float* __restrict__ root,
                                                    float* __restrict__ tmp) {
    int idx = blockIdx.x * 256 + threadIdx.x;   // N*32
    int v = idx >> 5, f = idx & 31;
    tmp[v * 64 + f] = fmaf(h[v * 64 + 0], root[f], h[v * 64 + 1] * root[32 + f]);
}

// hnext = ELU(tmp + segment_sum(msg over dst) + bias); one wave per vertex
template <int Fout>
__global__ __launch_bounds__(256) void finalize_kernel(const float* __restrict__ tmp,
                                                       const float* __restrict__ msg,
                                                       const float* __restrict__ bias,
                                                       const int* __restrict__ off,
                                                       const int* __restrict__ eid,
                                                       float* __restrict__ hnext) {
    int warp = threadIdx.x >> 5, lane = threadIdx.x & 31;
    int v = blockIdx.x * 8 + warp;
    float s0 = tmp[v * 64 + lane] + bias[lane];
    float s1 = (Fout == 64) ? (tmp[v * 64 + 32 + lane] + bias[32 + lane]) : 0.0f;
    int a = off[v], b = off[v + 1];
    for (int j = a; j < b; j++) {
        int e = eid[j];
        s0 += msg[(size_t)e * 64 + lane];
        if constexpr (Fout == 64) s1 += msg[(size_t)e * 64 + 32 + lane];
    }
    s0 = (s0 > 0.0f) ? s0 : (expf(s0) - 1.0f);
    hnext[v * 64 + lane] = s0;
    if constexpr (Fout == 64) {
        s1 = (s1 > 0.0f) ? s1 : (expf(s1) - 1.0f);
        hnext[v * 64 + 32 + lane] = s1;
    }
}

// layer-5 message (Fin=32, Fout=1): one thread per edge
__global__ __launch_bounds__(256) void edge_msg_f1_kernel(const float* __restrict__ h,
                                                          const float* __restrict__ W,
                                                          const int* __restrict__ ei,
                                                          const float* __restrict__ basis,
                                                          const int* __restrict__ wi,
                                                          float* __restrict__ msg) {
    int e = blockIdx.x * 256 + threadIdx.x;
    if (e >= NE) return;
    int src = ei[e];
    float acc = 0.0f;
    #pragma unroll
    for (int s = 0; s < 8; s++) {
        int k = wi[e * 8 + s];
        float bs = basis[e * 8 + s];
        const float* Wk = W + (size_t)k * 32;
        float d = 0.0f;
        #pragma unroll
        for (int i = 0; i < 32; i++) d = fmaf(h[src * 64 + i], Wk[i], d);
        acc = fmaf(bs, d, acc);
    }
    msg[e] = acc;
}

// layer-5 finalize: out[v] = ELU(sum msg + h.root + b)
__global__ __launch_bounds__(256) void final5_kernel(const float* __restrict__ h,
                                                     const float* __restrict__ root,
                                                     const float* __restrict__ bias,
                                                     const float* __restrict__ msg,
                                                     const int* __restrict__ off,
                                                     const int* __restrict__ eid,
                                                     float* __restrict__ out) {
    int v = blockIdx.x * 256 + threadIdx.x;
    if (v >= NV) return;
    float s = bias[0];
    #pragma unroll
    for (int i = 0; i < 32; i++) s = fmaf(h[v * 64 + i], root[i], s);
    int a = off[v], b = off[v + 1];
    for (int j = a; j < b; j++) s += msg[eid[j]];
    out[v] = (s > 0.0f) ? s : (expf(s) - 1.0f);
}

// ---------------- launch ----------------

extern "C" void kernel_launch(void* const* d_in, const int* in_sizes, int n_in,
                              void* d_out, int out_size, void* d_ws, size_t ws_size,
                              hipStream_t stream) {
    const float* x      = (const float*)d_in[0];
    const int*   ei     = (const int*)d_in[1];
    const float* pseudo = (const float*)d_in[2];
    const float *W[5], *root[5], *bias[5];
    for (int i = 0; i < 5; i++) {
        W[i]    = (const float*)d_in[3 + 3 * i];
        root[i] = (const float*)d_in[4 + 3 * i];
        bias[i] = (const float*)d_in[5 + 3 * i];
    }

    char* p = (char*)d_ws;
    size_t o = 0;
    auto alloc = [&](size_t bytes) -> void* {
        void* r = p + o;
        o = (o + bytes + 255) & ~(size_t)255;
        return r;
    };
    float* basis  = (float*)alloc((size_t)NE * 8 * 4);
    int*   wi     = (int*)  alloc((size_t)NE * 8 * 4);
    int*   counts = (int*)  alloc((size_t)NV * 4);
    int*   cursor = (int*)  alloc((size_t)NV * 4);
    int*   voff   = (int*)  alloc((size_t)(NV + 1) * 4);
    int*   eid    = (int*)  alloc((size_t)NE * 4);
    float* hA     = (float*)alloc((size_t)NV * 64 * 4);
    float* hB     = (float*)alloc((size_t)NV * 64 * 4);
    float* tmp    = (float*)alloc((size_t)NV * 64 * 4);
    float* msg    = (float*)alloc((size_t)NE * 64 * 4);

    // setup: transform + basis + deterministic CSR over dst
    zero_kernel<<<(2 * NV) / 256, 256, 0, stream>>>(counts, cursor);
    prep_kernel<<<NE / 256, 256, 0, stream>>>(x, ei, pseudo, basis, wi, hA, counts);
    scan_kernel<<<1, 256, 0, stream>>>(counts, voff);
    fill_csr_kernel<<<NE / 256, 256, 0, stream>>>(ei, voff, cursor, eid);
    sort_csr_kernel<<<NV / 256, 256, 0, stream>>>(voff, eid);

    // layer 1: (2 -> 32)
    edge_msg_kernel<2, 32><<<NE / 8, 256, 0, stream>>>(hA, W[0], ei, basis, wi, msg);
    root2_kernel<<<(NV * 32) / 256, 256, 0, stream>>>(hA, root[0], tmp);
    finalize_kernel<32><<<NV / 8, 256, 0, stream>>>(tmp, msg, bias[0], voff, eid, hB);

    // layer 2: (32 -> 64)
    edge_msg_kernel<32, 64><<<NE / 8, 256, 0, stream>>>(hB, W[1], ei, basis, wi, msg);
    wmma_root_kernel<32, 64><<<(512 * 4) / 8, 256, 0, stream>>>(hB, root[1], tmp);
    finalize_kernel<64><<<NV / 8, 256, 0, stream>>>(tmp, msg, bias[1], voff, eid, hA);

    // layer 3: (64 -> 64)
    edge_msg_kernel<64, 64><<<NE / 8, 256, 0, stream>>>(hA, W[2], ei, basis, wi, msg);
    wmma_root_kernel<64, 64><<<(512 * 4) / 8, 256, 0, stream>>>(hA, root[2], tmp);
    finalize_kernel<64><<<NV / 8, 256, 0, stream>>>(tmp, msg, bias[2], voff, eid, hB);

    // layer 4: (64 -> 32)
    edge_msg_kernel<64, 32><<<NE / 8, 256, 0, stream>>>(hB, W[3], ei, basis, wi, msg);
    wmma_root_kernel<64, 32><<<(512 * 2) / 8, 256, 0, stream>>>(hB, root[3], tmp);
    finalize_kernel<32><<<NV / 8, 256, 0, stream>>>(tmp, msg, bias[3], voff, eid, hA);

    // layer 5: (32 -> 1) -> d_out
    edge_msg_f1_kernel<<<NE / 256, 256, 0, stream>>>(hA, W[4], ei, basis, wi, msg);
    final5_kernel<<<NV / 256, 256, 0, stream>>>(hA, root[4], bias[4], msg, voff, eid,
                                                (float*)d_out);
}